// StealNMSLoss_40003325395430
// MI455X (gfx1250) — compile-verified
//
#include <hip/hip_runtime.h>
#include <math.h>

// Problem dims from reference setup_inputs(): B=4, C=19, H=512, W=512, fp32.
#define BN 4
#define CN 19
#define HN 512
#define WN 512
#define TS 32                 // output tile: 32x32 pixels per block
#define LTW 37                // padded row stride of 36-wide true tile
#define LEW 35                // padded row stride of 34-wide e tile
#define NBLK (BN * CN * (HN / TS) * (WN / TS))   // 76 * 256 = 19456

// CDNA5 async global->LDS staging (GLOBAL_LOAD_ASYNC_TO_LDS_B32, ASYNCcnt).
// Set to 0 to fall back to plain load+ds_store staging if asm is rejected.
#define USE_ASYNC_STAGE 1

static_assert((HN % TS) == 0 && (WN % TS) == 0, "tile must divide image");

__device__ __forceinline__ int clampi(int v, int lo, int hi) {
    return v < lo ? lo : (v > hi ? hi : v);
}

// Generic pointers to LDS on AMDGPU are {shared_aperture_hi32, lds_offset_lo32};
// the low 32 bits are the LDS byte address the async-load VDST operand wants.
__device__ __forceinline__ unsigned int lds_addr32(const void* p) {
    return (unsigned int)(unsigned long long)p;
}

typedef __attribute__((ext_vector_type(2))) float v2f;
typedef __attribute__((ext_vector_type(8))) float v8f;

#ifdef __has_builtin
#if __has_builtin(__builtin_amdgcn_wmma_f32_16x16x4_f32)
#define USE_WMMA_RED 1
#endif
#endif

__global__ __launch_bounds__(256)
void nms_loss_partial(const float* __restrict__ tru,
                      const float* __restrict__ prd,
                      float* __restrict__ partials)
{
    // LDS staging: clamp (edge) baked into true halo, reflect baked into e halo.
    __shared__ float LT[36 * LTW];     // true tile, halo 2 (5x5 effective stencil)
    __shared__ float LE[34 * LEW];     // exp(pred*TAU) tile, halo 1
    __shared__ float wsum[8];

    const int   tid   = threadIdx.x;
    const int   blk   = blockIdx.x;
    const int   tcx   = blk & 15;
    const int   tcy   = (blk >> 4) & 15;
    const int   plane = blk >> 8;                  // 0..75  (B*C planes)
    const int   ty0   = tcy * TS;
    const int   tx0   = tcx * TS;
    const float* tp   = tru + (size_t)plane * (HN * WN);
    const float* pp   = prd + (size_t)plane * (HN * WN);

    const float TAUc = 0.1f;
    const float EPSc = 1e-7f;

#if USE_ASYNC_STAGE
    // ---- async stage true tile with edge-clamped halo ----
    for (int i = tid; i < 36 * 36; i += 256) {
        int r  = i / 36, c = i % 36;
        int ys = clampi(ty0 - 2 + r, 0, HN - 1);
        int xs = clampi(tx0 - 2 + c, 0, WN - 1);
        const float* g = tp + ys * WN + xs;
        unsigned int  l = lds_addr32(&LT[r * LTW + c]);
        asm volatile("global_load_async_to_lds_b32 %0, %1, off"
                     :: "v"(l), "v"(g) : "memory");
    }
    // ---- async stage raw pred tile with reflect halo ----
    for (int i = tid; i < 34 * 34; i += 256) {
        int r = i / 34, c = i % 34;
        int q = ty0 - 1 + r;
        int ys = (q < 0) ? -q : ((q >= HN) ? (2 * HN - 2 - q) : q);
        q = tx0 - 1 + c;
        int xs = (q < 0) ? -q : ((q >= WN) ? (2 * WN - 2 - q) : q);
        const float* g = pp + ys * WN + xs;
        unsigned int  l = lds_addr32(&LE[r * LEW + c]);
        asm volatile("global_load_async_to_lds_b32 %0, %1, off"
                     :: "v"(l), "v"(g) : "memory");
    }
    // Wait for this wave's async transfers; each thread then exponentiates the
    // exact LE elements it loaded (same i-mapping), so no cross-wave hazard
    // before the barrier.
    asm volatile("s_wait_asynccnt 0" ::: "memory");
    for (int i = tid; i < 34 * 34; i += 256) {
        int r = i / 34, c = i % 34;
        LE[r * LEW + c] = expf(TAUc * LE[r * LEW + c]);
    }
#else
    // ---- fallback: load + ds_store staging ----
    for (int i = tid; i < 36 * 36; i += 256) {
        int r  = i / 36, c = i % 36;
        int ys = clampi(ty0 - 2 + r, 0, HN - 1);
        int xs = clampi(tx0 - 2 + c, 0, WN - 1);
        LT[r * LTW + c] = tp[ys * WN + xs];
    }
    for (int i = tid; i < 34 * 34; i += 256) {
        int r = i / 34, c = i % 34;
        int q = ty0 - 1 + r;
        int ys = (q < 0) ? -q : ((q >= HN) ? (2 * HN - 2 - q) : q);
        q = tx0 - 1 + c;
        int xs = (q < 0) ? -q : ((q >= WN) ? (2 * WN - 2 - q) : q);
        LE[r * LEW + c] = expf(TAUc * pp[ys * WN + xs]);
    }
#endif
    __syncthreads();

    // ---- per-pixel fused compute ----
    float acc = 0.f;
    const int lx  = tid & 31;
    const int ly0 = tid >> 5;

    #pragma unroll
    for (int k = 0; k < 4; ++k) {
        const int lyy = ly0 + 8 * k;         // row in tile 0..31
        const int y   = ty0 + lyy;
        const int x   = tx0 + lx;

        const float tv = LT[(lyy + 2) * LTW + (lx + 2)];
        if (tv > 0.f) {
            // gx, gy at the 3x3 of *clamped* coords (second conv's edge padding),
            // each reading a 3x3 true window whose own clamp is baked into LT.
            float gxv[9], gyv[9];
            #pragma unroll
            for (int d = 0; d < 9; ++d) {
                const int dy = d / 3 - 1, dx = d % 3 - 1;
                const int r2 = clampi(y + dy, 0, HN - 1) - ty0 + 2;  // LT row
                const int c2 = clampi(x + dx, 0, WN - 1) - tx0 + 2;  // LT col
                const float* p0 = &LT[(r2 - 1) * LTW + (c2 - 1)];
                const float t00 = p0[0],         t01 = p0[1],           t02 = p0[2];
                const float t10 = p0[LTW],                              t12 = p0[LTW + 2];
                const float t20 = p0[2 * LTW],   t21 = p0[2 * LTW + 1], t22 = p0[2 * LTW + 2];
                gxv[d] = ((t02 - t00) + 2.f * (t12 - t10) + (t22 - t20)) * 0.125f;
                gyv[d] = ((t20 - t00) + 2.f * (t21 - t01) + (t22 - t02)) * 0.125f;
            }
            // second-level cross-correlations (GX zero middle column, GY zero middle row)
            const float gxx = ((gxv[2] - gxv[0]) + 2.f * (gxv[5] - gxv[3]) + (gxv[8] - gxv[6])) * 0.125f;
            const float gxy = ((gyv[2] - gyv[0]) + 2.f * (gyv[5] - gyv[3]) + (gyv[8] - gyv[6])) * 0.125f;
            const float gyy = ((gyv[6] - gyv[0]) + 2.f * (gyv[7] - gyv[1]) + (gyv[8] - gyv[2])) * 0.125f;

            const float ns = -gxy + EPSc;
            const float sg = (ns > 0.f) ? 1.f : ((ns < 0.f) ? -1.f : 0.f);
            float angle = atanf(gyy * sg / (gxx + EPSc)) * 57.29577951308232f;
            if (angle < 0.f) angle += 180.f;

            // directional 3-tap sum of e with reflect baked into LE
            const int ci = (lyy + 1) * LEW + (lx + 1);
            const float ec = LE[ci];
            float f;
            bool have = true;
            if (angle < 22.5f || angle >= 157.5f) {       // horizontal
                f = LE[ci - 1] + ec + LE[ci + 1];
            } else if (angle < 67.5f) {                    // counter-diag kernel (fliplr eye)
                f = LE[ci - LEW + 1] + ec + LE[ci + LEW - 1];
            } else if (angle < 112.5f) {                   // vertical
                f = LE[ci - LEW] + ec + LE[ci + LEW];
            } else if (angle > 112.5f) {                   // leading diag (eye)
                f = LE[ci - LEW - 1] + ec + LE[ci + LEW + 1];
            } else {                                       // angle == 112.5 exactly: no band
                f = 0.f; have = false;
            }
            if (have) {
                float r = ec / (f + EPSc);
                r = fminf(fmaxf(r, EPSc), 1.0f);
                acc += logf(r);
            }
        }
    }

    // ---- block reduction: wave sums via V_WMMA_F32_16X16X4_F32 ----
#ifdef USE_WMMA_RED
    {
        // A (16x4): lane m -> A[m,0]=acc_m (lanes 0-15), A[m,2]=acc_{m+16}; K=1,3 zero.
        // B (4x16) = ones  =>  D[m,n] = acc_m + acc_{m+16}  (row sums, replicated in n).
        v2f a; a.x = acc; a.y = 0.f;
        v2f b; b.x = 1.f; b.y = 1.f;
        v8f c = {};
        v8f dm = __builtin_amdgcn_wmma_f32_16x16x4_f32(
            false, a, false, b, (short)0, c, false, false);
        float s8 = 0.f;
        #pragma unroll
        for (int i = 0; i < 8; ++i) s8 += dm[i];
        // wave total = (sum over M=0..7) from lane 0 + (sum over M=8..15) from lane 16
        const int sb = __float_as_int(s8);
        const float wtot = __int_as_float(__builtin_amdgcn_ds_bpermute(0, sb)) +
                           __int_as_float(__builtin_amdgcn_ds_bpermute(16 * 4, sb));
        if ((tid & 31) == 0) wsum[tid >> 5] = wtot;
    }
    __syncthreads();
    if (tid == 0) {
        float t = 0.f;
        #pragma unroll
        for (int i = 0; i < 8; ++i) t += wsum[i];
        partials[blk] = t;
    }
#else
    __syncthreads();
    LT[tid] = acc;                 // LT no longer needed; reuse as scratch
    __syncthreads();
    for (int s = 128; s > 0; s >>= 1) {
        if (tid < s) LT[tid] += LT[tid + s];
        __syncthreads();
    }
    if (tid == 0) partials[blk] = LT[0];
#endif
}

__global__ __launch_bounds__(256)
void nms_loss_reduce(const float* __restrict__ partials, int n, float* __restrict__ out)
{
    __shared__ float sm[256];
    float s = 0.f;
    for (int i = threadIdx.x; i < n; i += 256) s += partials[i];   // fixed order: deterministic
    sm[threadIdx.x] = s;
    __syncthreads();
    for (int k = 128; k > 0; k >>= 1) {
        if (threadIdx.x < k) sm[threadIdx.x] += sm[threadIdx.x + k];
        __syncthreads();
    }
    if (threadIdx.x == 0) out[0] = -sm[0] * (1.0f / (float)BN);
}

extern "C" void kernel_launch(void* const* d_in, const int* in_sizes, int n_in,
                              void* d_out, int out_size, void* d_ws, size_t ws_size,
                              hipStream_t stream) {
    const float* tru = (const float*)d_in[0];   // true_labels (4,19,512,512) f32
    const float* prd = (const float*)d_in[1];   // pred_labels (4,19,512,512) f32
    float* out      = (float*)d_out;            // scalar f32
    float* partials = (float*)d_ws;             // NBLK floats (~76 KB) of scratch

    nms_loss_partial<<<NBLK, 256, 0, stream>>>(tru, prd, partials);
    nms_loss_reduce<<<1, 256, 0, stream>>>(partials, NBLK, out);
}